// GCN5_20693152432416
// MI455X (gfx1250) — compile-verified
//
#include <hip/hip_runtime.h>
#include <hip/hip_bf16.h>

#define N_NODES  100000
#define N_EDGES  1000000
#define F_IN     128
#define HID      64
#define N_GRAPHS 1000
#define LN_EPS   1e-5f

typedef __attribute__((ext_vector_type(2))) float v2f;
typedef __attribute__((ext_vector_type(8))) float v8f;

// ---------------------------------------------------------------------------
// Degree / normalization: deg = 1 (self-loop) + #incoming edges; dis = rsqrt.
// ---------------------------------------------------------------------------
__global__ void k_deg_init(float* __restrict__ deg) {
    int i = blockIdx.x * blockDim.x + threadIdx.x;
    if (i < N_NODES) deg[i] = 1.0f;  // self-loop contribution
}

__global__ void k_deg_accum(const int* __restrict__ dst, float* __restrict__ deg) {
    int e = blockIdx.x * blockDim.x + threadIdx.x;
    if (e < N_EDGES) unsafeAtomicAdd(&deg[dst[e]], 1.0f);
}

__global__ void k_deg_rsqrt(float* __restrict__ deg) {
    int i = blockIdx.x * blockDim.x + threadIdx.x;
    if (i < N_NODES) deg[i] = rsqrtf(deg[i]);  // deg >= 1 always (self-loops)
}

// ---------------------------------------------------------------------------
// Dense GEMM via V_WMMA_F32_16X16X4_F32 (full fp32 matrix pipe).
// C[M x 64] = A[M x K] * B[K x 64].  One wave -> one 16x16 tile.
// Block = 128 threads = 4 waves -> 16 rows x full 64 cols. Grid = M/16.
// ---------------------------------------------------------------------------
__global__ void k_gemm_wmma(const float* __restrict__ A,
                            const float* __restrict__ B,
                            float* __restrict__ C, int K) {
    const int lane  = threadIdx.x & 31;
    const int wave  = threadIdx.x >> 5;       // 0..3: which 16-col tile
    const int rowB  = blockIdx.x << 4;
    const int colB  = wave << 4;
    const int lrow  = lane & 15;
    const int hi    = lane >> 4;              // 0: low half, 1: high half
    const int khalf = hi << 1;                // lanes 16-31 carry K+2, K+3

    const float* __restrict__ arow = A + (size_t)(rowB + lrow) * K;
    const float* __restrict__ bcol = B + colB + lrow;

    v8f acc = {};
    for (int k = 0; k < K; k += 4) {
        v2f a, b;
        a[0] = arow[k + khalf];
        a[1] = arow[k + khalf + 1];
        b[0] = bcol[(size_t)(k + khalf) * HID];
        b[1] = bcol[(size_t)(k + khalf + 1) * HID];
        acc = __builtin_amdgcn_wmma_f32_16x16x4_f32(
            /*neg_a=*/false, a, /*neg_b=*/false, b,
            /*c_mod=*/(short)0, acc, /*reuse_a=*/false, /*reuse_b=*/false);
    }

    // C/D layout: VGPR j -> row j (lanes 0-15) / row j+8 (lanes 16-31)
    float* crow = C + (size_t)(rowB + (hi << 3)) * HID + colB + lrow;
#pragma unroll
    for (int j = 0; j < 8; ++j) crow[(size_t)j * HID] = acc[j];
}

// ---------------------------------------------------------------------------
// agg = tmp * dis^2   (self-loop term folded into accumulator init)
// ---------------------------------------------------------------------------
__global__ void k_self_init(const float* __restrict__ tmp,
                            const float* __restrict__ dis,
                            float* __restrict__ agg) {
    size_t t = (size_t)blockIdx.x * blockDim.x + threadIdx.x;
    if (t >= (size_t)N_NODES * HID) return;
    int node = (int)(t >> 6);
    float d  = dis[node];
    agg[t]   = tmp[t] * d * d;
}

// ---------------------------------------------------------------------------
// Edge scatter: agg[dst] += tmp[src] * dis[src]*dis[dst].
// 16 threads per edge, one float4 each: coalesced 256B gather per edge,
// 4 native global_atomic_add_f32 per thread (L2-resident working set).
// ---------------------------------------------------------------------------
__global__ void k_scatter(const int* __restrict__ src,
                          const int* __restrict__ dst,
                          const float* __restrict__ dis,
                          const float* __restrict__ tmp,
                          float* __restrict__ agg) {
    size_t t = (size_t)blockIdx.x * blockDim.x + threadIdx.x;
    if (t >= (size_t)N_EDGES * 16) return;
    int e = (int)(t >> 4);
    int q = (int)(t & 15) << 2;
    int s = src[e], d = dst[e];
    float w = dis[s] * dis[d];
    const float4 v = *(const float4*)(tmp + (size_t)s * HID + q);
    float* o = agg + (size_t)d * HID + q;
    unsafeAtomicAdd(o + 0, v.x * w);
    unsafeAtomicAdd(o + 1, v.y * w);
    unsafeAtomicAdd(o + 2, v.z * w);
    unsafeAtomicAdd(o + 3, v.w * w);
}

__global__ void k_bias_relu(const float* __restrict__ agg,
                            const float* __restrict__ b,
                            float* __restrict__ h) {
    size_t t = (size_t)blockIdx.x * blockDim.x + threadIdx.x;
    if (t >= (size_t)N_NODES * HID) return;
    h[t] = fmaxf(agg[t] + b[t & 63], 0.0f);
}

// ---------------------------------------------------------------------------
// Mean pooling over graphs
// ---------------------------------------------------------------------------
__global__ void k_pool_zero(float* __restrict__ sums, float* __restrict__ cnt) {
    int i = blockIdx.x * blockDim.x + threadIdx.x;
    if (i < N_GRAPHS * HID) sums[i] = 0.0f;
    if (i < N_GRAPHS) cnt[i] = 0.0f;
}

__global__ void k_pool_accum(const float* __restrict__ h,
                             const int* __restrict__ batch,
                             float* __restrict__ sums,
                             float* __restrict__ cnt) {
    size_t t = (size_t)blockIdx.x * blockDim.x + threadIdx.x;
    if (t >= (size_t)N_NODES * 16) return;
    int node = (int)(t >> 4);
    int q    = (int)(t & 15) << 2;
    int g    = batch[node];
    const float4 v = *(const float4*)(h + (size_t)node * HID + q);
    float* o = sums + (size_t)g * HID + q;
    unsafeAtomicAdd(o + 0, v.x);
    unsafeAtomicAdd(o + 1, v.y);
    unsafeAtomicAdd(o + 2, v.z);
    unsafeAtomicAdd(o + 3, v.w);
    if (q == 0) unsafeAtomicAdd(&cnt[g], 1.0f);
}

// ---------------------------------------------------------------------------
// pooled = sums/cnt; LayerNorm over 64 feats; out = ln @ Wout + bout.
// One thread per graph (1000 threads total; trivially cheap).
// ---------------------------------------------------------------------------
__global__ void k_finalize(const float* __restrict__ sums,
                           const float* __restrict__ cnt,
                           const float* __restrict__ Wout,
                           const float* __restrict__ bout,
                           float* __restrict__ out) {
    int g = blockIdx.x * blockDim.x + threadIdx.x;
    if (g >= N_GRAPHS) return;
    float c   = fmaxf(cnt[g], 1.0f);
    float inv_c = 1.0f / c;
    const float* row = sums + (size_t)g * HID;

    float mu = 0.0f;
#pragma unroll
    for (int f = 0; f < HID; ++f) mu += row[f];
    mu = mu * inv_c * (1.0f / HID);

    float var = 0.0f;
#pragma unroll
    for (int f = 0; f < HID; ++f) {
        float p = row[f] * inv_c - mu;
        var += p * p;
    }
    var *= (1.0f / HID);
    float inv = rsqrtf(var + LN_EPS);

    float acc = 0.0f;
#pragma unroll
    for (int f = 0; f < HID; ++f)
        acc += (row[f] * inv_c - mu) * Wout[f];
    out[g] = acc * inv + bout[0];
}

// ---------------------------------------------------------------------------
extern "C" void kernel_launch(void* const* d_in, const int* in_sizes, int n_in,
                              void* d_out, int out_size, void* d_ws, size_t ws_size,
                              hipStream_t stream) {
    const float* x     = (const float*)d_in[0];
    const int*   ei    = (const int*)d_in[1];
    const int*   batch = (const int*)d_in[2];
    const float* W1    = (const float*)d_in[3];
    const float* b1    = (const float*)d_in[4];
    const float* Wh    = (const float*)d_in[5];
    const float* bh    = (const float*)d_in[6];
    const float* Wout  = (const float*)d_in[7];
    const float* bout  = (const float*)d_in[8];
    float* out = (float*)d_out;

    const int* src = ei;             // edge_index[0]
    const int* dst = ei + N_EDGES;   // edge_index[1]

    // workspace carve (floats); ~78 MB total — L2-resident on MI455X
    float* ws   = (float*)d_ws;
    float* dis  = ws;
    float* tmp  = dis + N_NODES;
    float* agg  = tmp + (size_t)N_NODES * HID;
    float* h    = agg + (size_t)N_NODES * HID;
    float* sums = h   + (size_t)N_NODES * HID;
    float* cnt  = sums + (size_t)N_GRAPHS * HID;

    const int T = 256;
    // deg -> dis
    k_deg_init  <<<(N_NODES + T - 1) / T, T, 0, stream>>>(dis);
    k_deg_accum <<<(N_EDGES + T - 1) / T, T, 0, stream>>>(dst, dis);
    k_deg_rsqrt <<<(N_NODES + T - 1) / T, T, 0, stream>>>(dis);

    const int nElem   = N_NODES * HID;            // 6.4M
    const int gElem   = (nElem + T - 1) / T;
    const int gScat   = (N_EDGES * 16 + T - 1) / T;

    for (int L = 0; L < 5; ++L) {
        const float* in = (L == 0) ? x : h;
        const int    K  = (L == 0) ? F_IN : HID;
        const float* W  = (L == 0) ? W1 : Wh + (size_t)(L - 1) * HID * HID;
        const float* b  = (L == 0) ? b1 : bh + (size_t)(L - 1) * HID;

        k_gemm_wmma <<<N_NODES / 16, 128, 0, stream>>>(in, W, tmp, K);
        k_self_init <<<gElem, T, 0, stream>>>(tmp, dis, agg);
        k_scatter   <<<gScat, T, 0, stream>>>(src, dst, dis, tmp, agg);
        k_bias_relu <<<gElem, T, 0, stream>>>(agg, b, h);
    }

    k_pool_zero  <<<(N_GRAPHS * HID + T - 1) / T, T, 0, stream>>>(sums, cnt);
    k_pool_accum <<<(N_NODES * 16 + T - 1) / T, T, 0, stream>>>(h, batch, sums, cnt);
    k_finalize   <<<(N_GRAPHS + 127) / 128, 128, 0, stream>>>(sums, cnt, Wout, bout, out);
}